// SingleMofifModel_84318797955784
// MI455X (gfx1250) — compile-verified
//
#include <hip/hip_runtime.h>
#include <hip/hip_bf16.h>

// Problem constants from the reference
// L=100, R=3, W0=5, W1=20, B=50
#define LL 100
#define RR 3
#define BB 50

typedef float v8f_t __attribute__((ext_vector_type(8)));
typedef float v2f_t __attribute__((ext_vector_type(2)));

#if __has_builtin(__builtin_amdgcn_global_load_async_to_lds_b32) && \
    __has_builtin(__builtin_amdgcn_s_wait_asynccnt)
#define HAVE_ASYNC_LDS 1
#else
#define HAVE_ASYNC_LDS 0
#endif

#if HAVE_ASYNC_LDS
typedef __attribute__((address_space(1))) int as1_int_t;
typedef __attribute__((address_space(3))) int as3_int_t;
#define ASYNC_G2L_B32(gptr, lptr)                                            \
    __builtin_amdgcn_global_load_async_to_lds_b32(                           \
        (as1_int_t*)(unsigned long long)(gptr), (as3_int_t*)(lptr), 0, 0)
#endif

__device__ __forceinline__ float sigm(float x) { return 1.0f / (1.0f + expf(-x)); }

__device__ __forceinline__ float mishf(float x) {
    float sp = (x > 20.f) ? x : log1pf(expf(x));   // softplus
    return x * tanhf(sp);
}

__device__ __forceinline__ float binomlp(float n, float k) {
    return lgammaf(n + 1.f) - lgammaf(k + 1.f) - lgammaf(n - k + 1.f) + n * logf(0.5f);
}

// Workspace layout (floats):
//   [0..239]     pwm_log (j*80 + k*4 + c)
//   [240..263]   w2f  (iw*3 + t)
//   [264..287]   w_prob (t*8 + iw)
//   [288..295]   wprod (iw)
//   [296]        base log_prior, [297] base log_guide
//   [300+blk*3]  per-block partials (prior, guide, logX), 4 blocks
//   [320..703]   gn_l1W packed  (Kpad=24, N=16, pair-interleaved)
//   [704..959]   gn_l2W packed  (Kpad=16, Npad 3->16)
//   [960..3775]  zn_clsW packed (Kpad=176, Npad 9->16)
//   [3776..6591] zn_delW packed (Kpad=176, Npad 10->16)
#define WS_L1   320
#define WS_L2   704
#define WS_CLS  960
#define WS_DEL  3776

// D += A(16xKpad, LDS row-major) * Bp(packed, zero-padded).
// Packed layout: Bp[(k>>1)*32 + n*2 + (k&1)] = B[k][n]  -> the (k,k+1) pair for
// column n is one aligned 8-byte load.
// A fragment per ISA 7.12.2 (32-bit A 16x4): lanes 0-15 hold K=k0,k0+1,
// lanes 16-31 hold K=k0+2,k0+3. C/D per the 16x16 f32 8-VGPR layout.
__device__ __forceinline__ void wmma_f32_k4(const float* A, int lda, int Kpad,
                                            const float* __restrict__ Bp,
                                            v8f_t& acc) {
    const int lane = (int)(threadIdx.x & 31u);
    const int half = lane >> 4;
    const int l15  = lane & 15;
    for (int k0 = 0; k0 < Kpad; k0 += 4) {
        const int ka = k0 + half * 2;          // even
        v2f_t a, b;
        a.x = A[l15 * lda + ka];
        a.y = A[l15 * lda + ka + 1];
        b   = *(const v2f_t*)(Bp + (ka >> 1) * 32 + l15 * 2);
        acc = __builtin_amdgcn_wmma_f32_16x16x4_f32(false, a, false, b,
                                                    (short)0, acc, false, false);
    }
}

// ---------------- Kernel A: pwm softmax + width network + weight packing ----
__global__ __launch_bounds__(32) void elbo_prep(
    const float* __restrict__ pwm_logits,
    const float* __restrict__ wn_W1, const float* __restrict__ wn_b1,
    const float* __restrict__ wn_W2, const float* __restrict__ wn_b2,
    const float* __restrict__ wn_W3, const float* __restrict__ wn_b3,
    const float* __restrict__ gn_l1W, const float* __restrict__ gn_l2W,
    const float* __restrict__ zn_clsW, const float* __restrict__ zn_delW,
    float* __restrict__ ws)
{
    __shared__ float red[32];
    const int tid = threadIdx.x;

    // ---- pack FC weights, zero-padded, pair-interleaved ----
    for (int idx = tid; idx < 24 * 16; idx += 32) {
        int k = idx >> 4, n = idx & 15;
        ws[WS_L1 + (k >> 1) * 32 + n * 2 + (k & 1)] = (k < 22) ? gn_l1W[k * 16 + n] : 0.f;
    }
    for (int idx = tid; idx < 16 * 16; idx += 32) {
        int k = idx >> 4, n = idx & 15;
        ws[WS_L2 + (k >> 1) * 32 + n * 2 + (k & 1)] = (n < 3) ? gn_l2W[k * 3 + n] : 0.f;
    }
    for (int idx = tid; idx < 176 * 16; idx += 32) {
        int k = idx >> 4, n = idx & 15;
        ws[WS_CLS + (k >> 1) * 32 + n * 2 + (k & 1)] = (n < 9)  ? zn_clsW[k * 9 + n]  : 0.f;
        ws[WS_DEL + (k >> 1) * 32 + n * 2 + (k & 1)] = (n < 10) ? zn_delW[k * 10 + n] : 0.f;
    }

    // ---- pwm softmax + Dirichlet prior ----
    const float dir4c = lgammaf(0.8f) - 4.f * lgammaf(0.2f);
    float dpart = 0.f;
    for (int row = tid; row < 60; row += 32) {          // 3*20 rows, K=4
        const float* lg = pwm_logits + row * 4;
        float m = fmaxf(fmaxf(lg[0], lg[1]), fmaxf(lg[2], lg[3]));
        float s = 0.f;
        for (int c = 0; c < 4; ++c) s += expf(lg[c] - m);
        float lse = m + logf(s);
        float lp = dir4c;
        for (int c = 0; c < 4; ++c) {
            float pl = lg[c] - lse;
            ws[row * 4 + c] = pl;                        // pwm_log
            lp += (0.2f - 1.f) * pl;
        }
        dpart += fminf(3.f, lp);
    }
    red[tid] = dpart;
    __syncthreads();
    if (tid == 0) {
        float dirsum = 0.f;
        for (int i = 0; i < 32; ++i) dirsum += red[i];   // fixed order

        float h1[16], h2[16], mean[3];
        for (int i = 0; i < 16; ++i) h1[i] = mishf(wn_W1[i] + wn_b1[i]);
        for (int j = 0; j < 16; ++j) {
            float a = wn_b2[j];
            for (int i = 0; i < 16; ++i) a += h1[i] * wn_W2[i * 16 + j];
            h2[j] = mishf(a);
        }
        for (int t = 0; t < 3; ++t) {
            float a = wn_b3[t];
            for (int j = 0; j < 16; ++j) a += h2[j] * wn_W3[j * 3 + t];
            mean[t] = sigm(a) * 15.f + 5.f;              // ww*(W1-W0)+W0
        }
        float w2f[8][3], wlp[8][3], wprod[8];
        for (int i = 0; i < 8; ++i) {
            wprod[i] = 1.f;
            for (int t = 0; t < 3; ++t) {
                int bit = (i >> (2 - t)) & 1;
                float v = bit ? ceilf(mean[t]) : floorf(mean[t]);
                w2f[i][t] = v;
                float up = (v >= 20.f) ? 1.f : sigm((v + 0.5f - mean[t]) / 0.05f);
                float lo = (v <= 5.f)  ? 0.f : sigm((v - 0.5f - mean[t]) / 0.05f);
                float p = up - lo + 1e-12f;
                wlp[i][t] = logf(p);
                ws[264 + t * 8 + i] = p;                 // w_prob[t][iw]
                wprod[i] *= p;
            }
        }
        float bprior = dirsum, bguide = 0.f;
        for (int i = 0; i < 8; ++i) {
            float bs = 0.f, ls = 0.f;
            for (int t = 0; t < 3; ++t) {
                bs += binomlp(15.f, w2f[i][t]);
                ls += wlp[i][t];
            }
            bprior += wprod[i] * bs;
            bguide += wprod[i] * ls;
            ws[288 + i] = wprod[i];
            for (int t = 0; t < 3; ++t) ws[240 + i * 3 + t] = w2f[i][t];
        }
        ws[296] = bprior;
        ws[297] = bguide;
    }
}

// ---------------- Kernel B: main per-batch-tile pipeline --------------------
// 4 blocks x 32 threads (one wave32/block); block covers batch rows [blk*16, +16)
__global__ __launch_bounds__(32) void elbo_main(
    const float* __restrict__ gn_c1W, const float* __restrict__ gn_c1b,
    const float* __restrict__ gn_c2W, const float* __restrict__ gn_c2b,
    const float* __restrict__ gn_l1b, const float* __restrict__ gn_l2b,
    const float* __restrict__ zn_c1W, const float* __restrict__ zn_c1b,
    const float* __restrict__ zn_c2W, const float* __restrict__ zn_c2b,
    const float* __restrict__ zn_clsb, const float* __restrict__ zn_delb,
    const int* __restrict__ X, float* __restrict__ ws)
{
    __shared__ float plog[240];      // pwm_log
    __shared__ float w2f_l[24];
    __shared__ float wprob_l[24];
    __shared__ float wprod_l[8];
    __shared__ int   Xl[16 * 100];
    __shared__ float yt[16 * 300];   // y[b][j][q]
    __shared__ float p1[16 * 48];    // pooled conv1 map (per-b scratch)
    __shared__ float g2pad[16 * 24]; // gn features, K padded 22->24
    __shared__ float hfc[16 * 16];   // gn fc1 activations
    __shared__ float Dt[16 * 16];    // WMMA D tile scratch
    __shared__ float gam[16 * 3];    // gamma
    __shared__ float zfeat[16 * 176];// zn features
    __shared__ float Zcls[16 * 16];
    __shared__ float Zdel[16 * 16];

    const int tid  = threadIdx.x;
    const int b0   = blockIdx.x * 16;
    const int lane = tid & 31, half = lane >> 4, l15 = lane & 15;

    __builtin_prefetch(X + b0 * 100, 0, 1);        // global_prefetch_b8
    __builtin_prefetch(ws + WS_CLS, 0, 1);
    __builtin_prefetch(ws + WS_DEL, 0, 1);

    // ---- stage pwm_log and X rows into LDS (async path when available) ----
#if HAVE_ASYNC_LDS
    for (int i = tid; i < 240; i += 32) {
        ASYNC_G2L_B32(ws + i, &plog[i]);
    }
    for (int i = tid; i < 1600; i += 32) {
        int b = i / 100, q = i % 100, bg = b0 + b;
        if (bg < BB) {
            ASYNC_G2L_B32(X + bg * 100 + q, &Xl[i]);
        } else {
            Xl[i] = 0;
        }
    }
    __builtin_amdgcn_s_wait_asynccnt(0);
#else
    for (int i = tid; i < 240; i += 32) plog[i] = ws[i];
    for (int i = tid; i < 1600; i += 32) {
        int b = i / 100, q = i % 100, bg = b0 + b;
        Xl[i] = (bg < BB) ? X[bg * 100 + q] : 0;
    }
#endif
    for (int i = tid; i < 24; i += 32) { w2f_l[i] = ws[240 + i]; wprob_l[i] = ws[264 + i]; }
    if (tid < 8) wprod_l[tid] = ws[288 + tid];
    __syncthreads();

    const float* Bl1  = ws + WS_L1;
    const float* Bl2  = ws + WS_L2;
    const float* Bcls = ws + WS_CLS;
    const float* Bdel = ws + WS_DEL;

    float acc_prior = 0.f, acc_guide = 0.f, acc_lX = 0.f;  // thread 0 owns these

    for (int iw = 0; iw < 8; ++iw) {
        int wint[3], lft[3];
        for (int t = 0; t < 3; ++t) {
            wint[t] = (int)w2f_l[iw * 3 + t];
            lft[t]  = (wint[t] - 1) >> 1;
        }
        // ---- y[b][j][q]: shifted windowed pwm scores ----
        for (int i = tid; i < 4800; i += 32) {
            int b = i / 300, r = i % 300, j = r / 100, q = r % 100;
            int bg = b0 + b, wj = wint[j], lj = lft[j];
            float val = 0.f;
            if (bg < BB && q >= lj && q <= lj + (LL - wj)) {
                int src = q - lj;
                float s = 0.f;
                for (int k = 0; k < wj; ++k) {
                    int pos = src + k; if (pos > 99) pos = 99;
                    s += plog[j * 80 + k * 4 + Xl[b * 100 + pos]];
                }
                val = s / (float)wj;
            }
            yt[i] = val;
        }
        for (int i = tid; i < 384; i += 32) g2pad[i] = 0.f;  // zero K-pad
        __syncthreads();

        // ---- gn: conv1(16,3,5)+mish+pool -> conv2(1,16,5)+mish+pool ----
        for (int b = 0; b < 16; ++b) {
            for (int i = tid; i < 768; i += 32) {
                int ch = i / 48, p2 = i % 48;
                float a0 = gn_c1b[ch], a1 = a0;
                for (int j = 0; j < 3; ++j)
                    for (int t = 0; t < 5; ++t) {
                        float w = gn_c1W[ch * 15 + j * 5 + t];
                        a0 += yt[b * 300 + j * 100 + 2 * p2 + t]     * w;
                        a1 += yt[b * 300 + j * 100 + 2 * p2 + 1 + t] * w;
                    }
                p1[i] = fmaxf(mishf(a0), mishf(a1));
            }
            __syncthreads();
            for (int p = tid; p < 22; p += 32) {
                float a0 = gn_c2b[0], a1 = a0;
                for (int ch = 0; ch < 16; ++ch)
                    for (int t = 0; t < 5; ++t) {
                        float w = gn_c2W[ch * 5 + t];
                        a0 += p1[ch * 48 + 2 * p + t]     * w;
                        a1 += p1[ch * 48 + 2 * p + 1 + t] * w;
                    }
                g2pad[b * 24 + p] = fmaxf(mishf(a0), mishf(a1));
            }
            __syncthreads();
        }
        // ---- gn fc1 (16x24)@(22x16) via WMMA, mish ----
        {
            v8f_t acc = {0.f,0.f,0.f,0.f,0.f,0.f,0.f,0.f};
            wmma_f32_k4(g2pad, 24, 24, Bl1, acc);
            #pragma unroll
            for (int r2 = 0; r2 < 8; ++r2)
                hfc[(r2 + half * 8) * 16 + l15] = mishf(acc[r2] + gn_l1b[l15]);
        }
        __syncthreads();
        // ---- gn fc2 (16x16)@(16x3) via WMMA, softmax -> gamma ----
        {
            v8f_t acc = {0.f,0.f,0.f,0.f,0.f,0.f,0.f,0.f};
            wmma_f32_k4(hfc, 16, 16, Bl2, acc);
            #pragma unroll
            for (int r2 = 0; r2 < 8; ++r2)
                Dt[(r2 + half * 8) * 16 + l15] = acc[r2];
        }
        __syncthreads();
        if (tid < 16) {
            float l0 = Dt[tid * 16 + 0] + gn_l2b[0];
            float l1 = Dt[tid * 16 + 1] + gn_l2b[1];
            float l2 = Dt[tid * 16 + 2] + gn_l2b[2];
            float m = fmaxf(l0, fmaxf(l1, l2));
            float e0 = expf(l0 - m), e1 = expf(l1 - m), e2 = expf(l2 - m);
            float s = e0 + e1 + e2;
            gam[tid * 3 + 0] = e0 / s; gam[tid * 3 + 1] = e1 / s; gam[tid * 3 + 2] = e2 / s;
        }
        __syncthreads();
        if (tid == 0) {
            const float dir3c = lgammaf(0.6f) - 3.f * lgammaf(0.2f);
            float ds = 0.f;
            for (int b = 0; b < 16; ++b) if (b0 + b < BB) {
                float lp = dir3c;
                for (int c = 0; c < 3; ++c) lp += (0.2f - 1.f) * logf(gam[b * 3 + c]);
                ds += fminf(3.f, lp);
            }
            acc_prior += wprod_l[iw] * ds;
        }
        __syncthreads();

        // ---- zn inner loop over motif I ----
        for (int I = 0; I < RR; ++I) {
            for (int b = 0; b < 16; ++b) {
                for (int i = tid; i < 768; i += 32) {
                    int ch = i / 48, p2 = i % 48;
                    float a0 = zn_c1b[ch], a1 = a0;
                    for (int t = 0; t < 5; ++t) {
                        float w = zn_c1W[ch * 5 + t];
                        a0 += yt[b * 300 + I * 100 + 2 * p2 + t]     * w;
                        a1 += yt[b * 300 + I * 100 + 2 * p2 + 1 + t] * w;
                    }
                    p1[i] = fmaxf(mishf(a0), mishf(a1));
                }
                __syncthreads();
                for (int i = tid; i < 176; i += 32) {
                    int oc = i / 22, p2 = i % 22;
                    float a0 = zn_c2b[oc], a1 = a0;
                    for (int ch = 0; ch < 16; ++ch)
                        for (int t = 0; t < 5; ++t) {
                            float w = zn_c2W[oc * 80 + ch * 5 + t];
                            a0 += p1[ch * 48 + 2 * p2 + t]     * w;
                            a1 += p1[ch * 48 + 2 * p2 + 1 + t] * w;
                        }
                    zfeat[b * 176 + oc * 22 + p2] = fmaxf(mishf(a0), mishf(a1));
                }
                __syncthreads();
            }
            // Z_cls = z @ (176x9) + b ; Z_del = sigmoid(z @ (176x10) + b)
            {
                v8f_t acc = {0.f,0.f,0.f,0.f,0.f,0.f,0.f,0.f};
                wmma_f32_k4(zfeat, 176, 176, Bcls, acc);
                #pragma unroll
                for (int r2 = 0; r2 < 8; ++r2)
                    Zcls[(r2 + half * 8) * 16 + l15] =
                        acc[r2] + (l15 < 9 ? zn_clsb[l15] : 0.f);
            }
            {
                v8f_t acc = {0.f,0.f,0.f,0.f,0.f,0.f,0.f,0.f};
                wmma_f32_k4(zfeat, 176, 176, Bdel, acc);
                #pragma unroll
                for (int r2 = 0; r2 < 8; ++r2)
                    Zdel[(r2 + half * 8) * 16 + l15] =
                        sigm(acc[r2] + (l15 < 10 ? zn_delb[l15] : 0.f));
            }
            __syncthreads();
            if (tid == 0) {
                const float wpI = wprob_l[I * 8 + iw];
                const int   wI  = wint[I];
                for (int b = 0; b < 16; ++b) if (b0 + b < BB) {
                    int ind = 0; float best = Zcls[b * 16 + 0];
                    for (int c = 1; c < 9; ++c) {
                        float v = Zcls[b * 16 + c];
                        if (v > best) { best = v; ind = c; }
                    }
                    ind += 1;                                  // 1..9
                    float zd  = Zdel[b * 16 + ind];
                    float Zm  = ((float)ind + zd) * 7.9f;      // gap=(99-W1)/10
                    float Ipr = gam[b * 3 + I];
                    float zf0 = floorf(Zm), zf1 = ceilf(Zm);
                    for (int s2 = 0; s2 < 2; ++s2) {
                        float zf = s2 ? zf1 : zf0;
                        int   zi = (int)zf;
                        float up = (zf >= 99.f) ? 1.f : sigm((zf + 0.5f - Zm) / 0.05f);
                        float lo = (zf <= 0.f)  ? 0.f : sigm((zf - 0.5f - Zm) / 0.05f);
                        float p  = up - lo + 1e-12f;
                        float zlp = logf(p);
                        float wzi = wpI * p * Ipr;
                        acc_prior += wzi * binomlp(79.f, zf);  // L-W1-1 = 79
                        acc_guide += wzi * zlp;
                        float lpv = 0.f;
                        for (int k = 0; k < wI; ++k) {
                            int pos = zi + k;
                            if (pos < 0) pos = 0; if (pos > 99) pos = 99;
                            lpv += plog[I * 80 + k * 4 + Xl[b * 100 + pos]];
                        }
                        acc_lX += wzi * lpv;
                    }
                }
            }
            __syncthreads();
        } // I
    } // iw

    if (tid == 0) {
        ws[300 + blockIdx.x * 3 + 0] = acc_prior;
        ws[300 + blockIdx.x * 3 + 1] = acc_guide;
        ws[300 + blockIdx.x * 3 + 2] = acc_lX;
    }
}

// ---------------- Kernel C: deterministic final reduction -------------------
__global__ void elbo_final(const float* __restrict__ ws, float* __restrict__ out) {
    float prior = ws[296], guide = ws[297], lX = 0.f;
    for (int blk = 0; blk < 4; ++blk) {
        prior += ws[300 + blk * 3 + 0];
        guide += ws[300 + blk * 3 + 1];
        lX    += ws[300 + blk * 3 + 2];
    }
    out[0] = -lX - prior + guide;
}

extern "C" void kernel_launch(void* const* d_in, const int* in_sizes, int n_in,
                              void* d_out, int out_size, void* d_ws, size_t ws_size,
                              hipStream_t stream) {
    (void)in_sizes; (void)n_in; (void)out_size; (void)ws_size;
    const float* pwm_logits = (const float*)d_in[0];
    const float* wn_W1  = (const float*)d_in[1];
    const float* wn_b1  = (const float*)d_in[2];
    const float* wn_W2  = (const float*)d_in[3];
    const float* wn_b2  = (const float*)d_in[4];
    const float* wn_W3  = (const float*)d_in[5];
    const float* wn_b3  = (const float*)d_in[6];
    const float* gn_c1W = (const float*)d_in[7];
    const float* gn_c1b = (const float*)d_in[8];
    const float* gn_c2W = (const float*)d_in[9];
    const float* gn_c2b = (const float*)d_in[10];
    const float* gn_l1W = (const float*)d_in[11];
    const float* gn_l1b = (const float*)d_in[12];
    const float* gn_l2W = (const float*)d_in[13];
    const float* gn_l2b = (const float*)d_in[14];
    const float* zn_c1W = (const float*)d_in[15];
    const float* zn_c1b = (const float*)d_in[16];
    const float* zn_c2W = (const float*)d_in[17];
    const float* zn_c2b = (const float*)d_in[18];
    const float* zn_clsW = (const float*)d_in[19];
    const float* zn_clsb = (const float*)d_in[20];
    const float* zn_delW = (const float*)d_in[21];
    const float* zn_delb = (const float*)d_in[22];
    const int*   X       = (const int*)d_in[23];
    float* ws  = (float*)d_ws;
    float* out = (float*)d_out;

    elbo_prep<<<1, 32, 0, stream>>>(pwm_logits, wn_W1, wn_b1, wn_W2, wn_b2,
                                    wn_W3, wn_b3, gn_l1W, gn_l2W,
                                    zn_clsW, zn_delW, ws);
    elbo_main<<<4, 32, 0, stream>>>(gn_c1W, gn_c1b, gn_c2W, gn_c2b,
                                    gn_l1b, gn_l2b,
                                    zn_c1W, zn_c1b, zn_c2W, zn_c2b,
                                    zn_clsb, zn_delb,
                                    X, ws);
    elbo_final<<<1, 1, 0, stream>>>(ws, out);
}